// DenseGrid_31009663877353
// MI455X (gfx1250) — compile-verified
//
#include <hip/hip_runtime.h>
#include <stdint.h>

// ---------------------------------------------------------------------------
// NGLOD multi-LOD bilinear feature interpolation for MI455X (gfx1250).
//  - LODs 0..2 (res 16/32/64, 84KB total) staged into LDS via the Tensor Data
//    Mover (TENSOR_LOAD_TO_LDS, TENSORcnt-tracked), served by ds_load_b128.
//  - LODs 3..7 gathered from global (L2-resident, 85.5MB < 192MB L2) with
//    global_load_b128.
//  - Output (256MB stream) written with non-temporal b128 stores so it does
//    not evict the grids from L2.
// ---------------------------------------------------------------------------

typedef __attribute__((ext_vector_type(4))) float        f4;
typedef __attribute__((ext_vector_type(2))) float        f2;
typedef __attribute__((ext_vector_type(4))) unsigned int u32x4;
typedef __attribute__((ext_vector_type(8))) int          i32x8;
typedef __attribute__((ext_vector_type(4))) int          i32x4;

#define BASE_LOD 4
#define NUM_LODS 8
#define FEAT_DIM 4

// LDS float offsets for staged grids (res 16, 32, 64)
#define SG0_OFF 0          // 16*16*4   = 1024 floats (4 KB)
#define SG1_OFF 1024       // 32*32*4   = 4096 floats (16 KB)
#define SG2_OFF 5120       // 64*64*4   = 16384 floats (64 KB)
#define SG_TOTAL 21504     // 86016 bytes

// Issue one TDM descriptor: copy a res x (res*4 floats) row-major 2D tile
// from global memory to LDS.  Layout per CDNA5 ISA.
__device__ __forceinline__ void tdm_load_grid(uint32_t lds_byte_off,
                                              const float* gptr,
                                              uint32_t res) {
  const uint32_t W = res * FEAT_DIM;            // floats per row (data_size=4B units)
  const uint64_t ga = (uint64_t)(uintptr_t)gptr;

  u32x4 g0;
  g0.x = 1u;                                    // count=1, user mode, no gather
  g0.y = lds_byte_off;                          // lds_addr (bytes)
  g0.z = (uint32_t)ga;                          // global_addr[31:0]
  g0.w = (uint32_t)(ga >> 32) | (2u << 30);     // global_addr[56:32] | type=2

  i32x8 g1;
  g1[0] = (int)(2u << 16);                      // wg_mask=0, data_size=2 (4B)
  g1[1] = (int)((W & 0xFFFFu) << 16);           // tensor_dim0[15:0] @bits63:48
  g1[2] = (int)(((W >> 16) & 0xFFFFu)           // tensor_dim0[31:16]
              | ((res & 0xFFFFu) << 16));       // tensor_dim1[15:0] @bits95:80
  g1[3] = (int)(((res >> 16) & 0xFFFFu)         // tensor_dim1[31:16]
              | ((W & 0xFFFFu) << 16));         // tile_dim0 @bits127:112
  g1[4] = (int)(res & 0xFFFFu);                 // tile_dim1 (rows); tile_dim2=0
  g1[5] = (int)W;                               // tensor_dim0_stride[31:0]
  g1[6] = 0;                                    // stride[47:32]=0
  g1[7] = 0;

  i32x4 gz4 = {0, 0, 0, 0};                     // groups 2/3 unused (<=2D tensor)
  i32x8 gz8 = {0, 0, 0, 0, 0, 0, 0, 0};
  __builtin_amdgcn_tensor_load_to_lds(g0, g1, gz4, gz4, gz8, 0);
}

// Exact reproduction of the reference bilerp math (clip -> floor for corner
// indices, weights computed from the *unclipped* scaled coordinate).
__device__ __forceinline__ f4 bilerp(const f4* __restrict__ g, int res,
                                     float px, float py) {
  const float fr = (float)(res - 1);
  const float x = px * fr;
  const float y = py * fr;
  const float hi = fr - 1e-5f;
  const float xc = fminf(fmaxf(x, 0.0f), hi);
  const float yc = fminf(fmaxf(y, 0.0f), hi);
  const int x1 = (int)xc;                       // xc >= 0 -> trunc == floor
  const int y1 = (int)yc;
  int x2 = x1 + 1; if (x2 > res - 1) x2 = res - 1;
  int y2 = y1 + 1; if (y2 > res - 1) y2 = res - 1;
  const float x1f = (float)x1, y1f = (float)y1;
  const float x2f = (float)x2, y2f = (float)y2;
  const float w1 = (x2f - x) * (y2f - y);
  const float w2 = (x - x1f) * (y2f - y);
  const float w3 = (x2f - x) * (y - y1f);
  const float w4 = (x - x1f) * (y - y1f);
  // Four 16B gathers (b128); issued together for memory-level parallelism.
  const f4 a = g[x1 + y1 * res];
  const f4 b = g[x2 + y1 * res];
  const f4 c = g[x1 + y2 * res];
  const f4 d = g[x2 + y2 * res];
  return w1 * a + w2 * b + w3 * c + w4 * d;
}

__global__ __launch_bounds__(256) void nglod_bilerp_kernel(
    const float* __restrict__ xin,
    const float* __restrict__ g0, const float* __restrict__ g1,
    const float* __restrict__ g2, const float* __restrict__ g3,
    const float* __restrict__ g4, const float* __restrict__ g5,
    const float* __restrict__ g6, const float* __restrict__ g7,
    float* __restrict__ out, int N) {
  __shared__ __align__(16) float sg[SG_TOTAL];

  // Wave 0 DMAs the three smallest grids into LDS.  TDM ignores EXEC but the
  // instruction issues per-wave, so branching the other waves around it is
  // correct; TENSORcnt is per-wave, so only wave 0 waits before the barrier.
  if (threadIdx.x < 32) {
    tdm_load_grid((uint32_t)(uintptr_t)&sg[SG0_OFF], g0, 16);
    tdm_load_grid((uint32_t)(uintptr_t)&sg[SG1_OFF], g1, 32);
    tdm_load_grid((uint32_t)(uintptr_t)&sg[SG2_OFF], g2, 64);
    __builtin_amdgcn_s_wait_tensorcnt(0);
  }
  __syncthreads();

  const int n = blockIdx.x * blockDim.x + threadIdx.x;
  if (n >= N) return;

  const f2 p = __builtin_nontemporal_load((const f2*)xin + n);  // streamed once
  const float px = p.x, py = p.y;

  f4* o = (f4*)out + (size_t)n * NUM_LODS;

  f4 r;
  // LDS-resident LODs (ds_load_b128 gathers)
  r = bilerp((const f4*)&sg[SG0_OFF],   16, px, py); __builtin_nontemporal_store(r, o + 0);
  r = bilerp((const f4*)&sg[SG1_OFF],   32, px, py); __builtin_nontemporal_store(r, o + 1);
  r = bilerp((const f4*)&sg[SG2_OFF],   64, px, py); __builtin_nontemporal_store(r, o + 2);
  // L2-resident LODs (global_load_b128 gathers, regular temporal hint)
  r = bilerp((const f4*)g3,  128, px, py); __builtin_nontemporal_store(r, o + 3);
  r = bilerp((const f4*)g4,  256, px, py); __builtin_nontemporal_store(r, o + 4);
  r = bilerp((const f4*)g5,  512, px, py); __builtin_nontemporal_store(r, o + 5);
  r = bilerp((const f4*)g6, 1024, px, py); __builtin_nontemporal_store(r, o + 6);
  r = bilerp((const f4*)g7, 2048, px, py); __builtin_nontemporal_store(r, o + 7);
}

extern "C" void kernel_launch(void* const* d_in, const int* in_sizes, int n_in,
                              void* d_out, int out_size, void* d_ws, size_t ws_size,
                              hipStream_t stream) {
  const float* xin = (const float*)d_in[0];
  const float* g0 = (const float*)d_in[1];
  const float* g1 = (const float*)d_in[2];
  const float* g2 = (const float*)d_in[3];
  const float* g3 = (const float*)d_in[4];
  const float* g4 = (const float*)d_in[5];
  const float* g5 = (const float*)d_in[6];
  const float* g6 = (const float*)d_in[7];
  const float* g7 = (const float*)d_in[8];

  const int N = in_sizes[0] / 2;                 // [N,2] coords
  const int threads = 256;
  const int blocks = (N + threads - 1) / threads;

  nglod_bilerp_kernel<<<blocks, threads, 0, stream>>>(
      xin, g0, g1, g2, g3, g4, g5, g6, g7, (float*)d_out, N);
}